// H_GCL_55997783605351
// MI455X (gfx1250) — compile-verified
//
#include <hip/hip_runtime.h>
#include <hip/hip_bf16.h>

// ---------------- problem constants (from reference) ----------------
#define N_NODES 8192
#define IN_F    512
#define HID_F   256
#define OUT_F   128
#define N_EDGES 262144
#define T_DIFF  20
#define GAMMA   0.8f
#define TEMP_INV (1.0f / 0.7f)
#define LN_EPS  1e-5f
#define CHUNKS  8
#define CHUNK_COLS (N_NODES / CHUNKS)   // 1024

typedef float v2f __attribute__((ext_vector_type(2)));
typedef float v8f __attribute__((ext_vector_type(8)));

// =====================================================================
// Elementwise / setup kernels
// =====================================================================
__global__ void k_zero(float* __restrict__ p, long n) {
  long i = (long)blockIdx.x * blockDim.x + threadIdx.x;
  long stride = (long)gridDim.x * blockDim.x;
  for (; i < n; i += stride) p[i] = 0.0f;
}

__global__ void k_mask_x(const float* __restrict__ x, const float* __restrict__ fu,
                         float* __restrict__ xb, long n) {
  long i = (long)blockIdx.x * blockDim.x + threadIdx.x;
  if (i < n) xb[i] = x[i] * (fu[i] > 0.3f ? 1.0f : 0.0f);
}

__global__ void k_edge_w(const float* __restrict__ eu, float* __restrict__ ew, int E) {
  int e = blockIdx.x * blockDim.x + threadIdx.x;
  if (e < E) ew[e] = (eu[e] > 0.2f) ? 1.0f : 0.0f;
}

__global__ void k_deg_gcn(const int* __restrict__ dst, const float* __restrict__ ew,
                          float* __restrict__ deg, int E) {
  int e = blockIdx.x * blockDim.x + threadIdx.x;
  if (e < E) atomicAdd(&deg[dst[e]], ew[e]);
}

__global__ void k_dinv(const float* __restrict__ deg, float* __restrict__ dinv, int n) {
  int i = blockIdx.x * blockDim.x + threadIdx.x;
  if (i < n) dinv[i] = rsqrtf(deg[i] + 1.0f);
}

__global__ void k_coef(const int* __restrict__ src, const int* __restrict__ dst,
                       const float* __restrict__ ew, const float* __restrict__ dinv,
                       float* __restrict__ coef, int E) {
  int e = blockIdx.x * blockDim.x + threadIdx.x;
  if (e < E) coef[e] = dinv[src[e]] * dinv[dst[e]] * ew[e];
}

__global__ void k_deg_hyper(const int* __restrict__ nidx, const int* __restrict__ eidx,
                            float* __restrict__ Dcnt, float* __restrict__ Bcnt, int E) {
  int e = blockIdx.x * blockDim.x + threadIdx.x;
  if (e < E) {
    atomicAdd(&Bcnt[eidx[e]], 1.0f);
    atomicAdd(&Dcnt[nidx[e]], 1.0f);
  }
}

__global__ void k_recip(float* __restrict__ p, int n) {
  int i = blockIdx.x * blockDim.x + threadIdx.x;
  if (i < n) { float v = p[i]; p[i] = (v > 0.0f) ? (1.0f / v) : 0.0f; }
}

__global__ void k_avg(const float* __restrict__ a, const float* __restrict__ b,
                      float* __restrict__ o, long n) {
  long i = (long)blockIdx.x * blockDim.x + threadIdx.x;
  if (i < n) o[i] = (a[i] + b[i]) * 0.5f;
}

__global__ void k_copy(const float* __restrict__ s, float* __restrict__ d, long n) {
  long i = (long)blockIdx.x * blockDim.x + threadIdx.x;
  if (i < n) d[i] = s[i];
}

// =====================================================================
// FP32 WMMA GEMM: C[M,Nc] = A[M,K] @ W[K,Nc] (+bias)(+relu)
// One wave per 16x16 tile; V_WMMA_F32_16X16X4_F32 over K in steps of 4.
// =====================================================================
template<int BIAS, int RELU>
__global__ __launch_bounds__(256)
void k_gemm_wmma(const float* __restrict__ A, const float* __restrict__ W,
                 const float* __restrict__ bias, float* __restrict__ C,
                 int M, int K, int Nc) {
  const int wave = (blockIdx.x * blockDim.x + threadIdx.x) >> 5;
  const int lane = threadIdx.x & 31;
  const int colTiles = Nc >> 4;
  const int row0 = (wave / colTiles) << 4;
  const int col0 = (wave % colTiles) << 4;
  if (row0 >= M) return;
  const int half = lane >> 4;
  const int l    = lane & 15;

  // A frag: lanes 0-15 -> K={0,1}; lanes 16-31 -> K={2,3}; M = lane&15
  const float* Arow = A + (size_t)(row0 + l) * K + 2 * half;
  // B frag: mirrored (K = 2*half + v, N = lane&15) -- lane-contiguous loads
  const float* Bp = W + (size_t)(2 * half) * Nc + col0 + l;

  v8f acc = {};
#pragma unroll 4
  for (int k0 = 0; k0 < K; k0 += 4) {
    float2 av = *(const float2*)(Arow + k0);
    v2f a; a.x = av.x; a.y = av.y;
    v2f b;
    b.x = Bp[(size_t)k0 * Nc];
    b.y = Bp[(size_t)k0 * Nc + Nc];
    acc = __builtin_amdgcn_wmma_f32_16x16x4_f32(false, a, false, b,
                                                (short)0, acc, false, false);
  }

  float bj = 0.0f;
  if (BIAS) bj = bias[col0 + l];
#pragma unroll
  for (int r = 0; r < 8; ++r) {
    float v = acc[r] + bj;
    if (RELU) v = fmaxf(v, 0.0f);
    C[(size_t)(row0 + r + 8 * half) * Nc + col0 + l] = v;
  }
}

// =====================================================================
// Scatter (segment-sum) kernels: one thread handles 4 features of 1 edge
// =====================================================================
__global__ void k_scatter_gcn(const float* __restrict__ xw, const int* __restrict__ src,
                              const int* __restrict__ dst, const float* __restrict__ coef,
                              float* __restrict__ agg, int E, int F) {
  long idx = (long)blockIdx.x * blockDim.x + threadIdx.x;
  int perEdge = F >> 2;
  long e = idx / perEdge;
  if (e >= E) return;
  int j = (int)(idx - e * perEdge) << 2;
  float c = coef[e];
  if (c == 0.0f) return;
  const float* xr = xw + (size_t)src[e] * F + j;
  float* ar = agg + (size_t)dst[e] * F + j;
  float4 v = *(const float4*)xr;
  atomicAdd(ar + 0, v.x * c); atomicAdd(ar + 1, v.y * c);
  atomicAdd(ar + 2, v.z * c); atomicAdd(ar + 3, v.w * c);
}

__global__ void k_scatter_n2e(const float* __restrict__ xw, const int* __restrict__ nidx,
                              const int* __restrict__ eidx, float* __restrict__ he,
                              int E, int F) {
  long idx = (long)blockIdx.x * blockDim.x + threadIdx.x;
  int perEdge = F >> 2;
  long e = idx / perEdge;
  if (e >= E) return;
  int j = (int)(idx - e * perEdge) << 2;
  const float* xr = xw + (size_t)nidx[e] * F + j;
  float* hr = he + (size_t)eidx[e] * F + j;
  float4 v = *(const float4*)xr;
  atomicAdd(hr + 0, v.x); atomicAdd(hr + 1, v.y);
  atomicAdd(hr + 2, v.z); atomicAdd(hr + 3, v.w);
}

__global__ void k_scatter_e2n(const float* __restrict__ he, const int* __restrict__ nidx,
                              const int* __restrict__ eidx, const float* __restrict__ Binv,
                              float* __restrict__ agg, int E, int F) {
  long idx = (long)blockIdx.x * blockDim.x + threadIdx.x;
  int perEdge = F >> 2;
  long e = idx / perEdge;
  if (e >= E) return;
  int j = (int)(idx - e * perEdge) << 2;
  float c = Binv[eidx[e]];
  if (c == 0.0f) return;
  const float* hr = he + (size_t)eidx[e] * F + j;
  float* ar = agg + (size_t)nidx[e] * F + j;
  float4 v = *(const float4*)hr;
  atomicAdd(ar + 0, v.x * c); atomicAdd(ar + 1, v.y * c);
  atomicAdd(ar + 2, v.z * c); atomicAdd(ar + 3, v.w * c);
}

// =====================================================================
// GCN epilogue: h = agg + xw*dinv^2 + bias; optional LayerNorm*g+b + ReLU
// One wave per row; feature-dim reductions via shuffles.
// =====================================================================
template<int F, int LNRELU>
__global__ __launch_bounds__(256)
void k_gcn_post(const float* __restrict__ agg, const float* __restrict__ xw,
                const float* __restrict__ dinv, const float* __restrict__ bias,
                const float* __restrict__ g, const float* __restrict__ be,
                float* __restrict__ out, int n) {
  int row  = (blockIdx.x * blockDim.x + threadIdx.x) >> 5;
  int lane = threadIdx.x & 31;
  if (row >= n) return;
  float d = dinv[row]; float d2 = d * d;
  const float* ar = agg + (size_t)row * F;
  const float* xr = xw  + (size_t)row * F;
  float vals[F / 32];
#pragma unroll
  for (int t = 0; t < F / 32; ++t) {
    int j = lane + t * 32;
    vals[t] = ar[j] + xr[j] * d2 + bias[j];
  }
  if (!LNRELU) {
#pragma unroll
    for (int t = 0; t < F / 32; ++t) out[(size_t)row * F + lane + t * 32] = vals[t];
    return;
  }
  float s = 0.0f;
#pragma unroll
  for (int t = 0; t < F / 32; ++t) s += vals[t];
#pragma unroll
  for (int off = 16; off >= 1; off >>= 1) s += __shfl_xor(s, off, 32);
  float mu = s / (float)F;
  float v2 = 0.0f;
#pragma unroll
  for (int t = 0; t < F / 32; ++t) { float dd = vals[t] - mu; v2 += dd * dd; }
#pragma unroll
  for (int off = 16; off >= 1; off >>= 1) v2 += __shfl_xor(v2, off, 32);
  float rstd = rsqrtf(v2 / (float)F + LN_EPS);
#pragma unroll
  for (int t = 0; t < F / 32; ++t) {
    int j = lane + t * 32;
    out[(size_t)row * F + j] = fmaxf((vals[t] - mu) * rstd * g[j] + be[j], 0.0f);
  }
}

// Hypergraph epilogue: h = relu(LN(agg*Dinv + bias)*g + b)
template<int F>
__global__ __launch_bounds__(256)
void k_hgc_post(const float* __restrict__ agg, const float* __restrict__ Dinv,
                const float* __restrict__ bias, const float* __restrict__ g,
                const float* __restrict__ be, float* __restrict__ out, int n) {
  int row  = (blockIdx.x * blockDim.x + threadIdx.x) >> 5;
  int lane = threadIdx.x & 31;
  if (row >= n) return;
  float di = Dinv[row];
  const float* ar = agg + (size_t)row * F;
  float vals[F / 32];
#pragma unroll
  for (int t = 0; t < F / 32; ++t) {
    int j = lane + t * 32;
    vals[t] = ar[j] * di + bias[j];
  }
  float s = 0.0f;
#pragma unroll
  for (int t = 0; t < F / 32; ++t) s += vals[t];
#pragma unroll
  for (int off = 16; off >= 1; off >>= 1) s += __shfl_xor(s, off, 32);
  float mu = s / (float)F;
  float v2 = 0.0f;
#pragma unroll
  for (int t = 0; t < F / 32; ++t) { float dd = vals[t] - mu; v2 += dd * dd; }
#pragma unroll
  for (int off = 16; off >= 1; off >>= 1) v2 += __shfl_xor(v2, off, 32);
  float rstd = rsqrtf(v2 / (float)F + LN_EPS);
#pragma unroll
  for (int t = 0; t < F / 32; ++t) {
    int j = lane + t * 32;
    out[(size_t)row * F + j] = fmaxf((vals[t] - mu) * rstd * g[j] + be[j], 0.0f);
  }
}

// =====================================================================
// Contrastive loss pieces
// =====================================================================
__global__ __launch_bounds__(256)
void k_rownorm(const float* __restrict__ h, float* __restrict__ z, int n) {
  int row  = (blockIdx.x * blockDim.x + threadIdx.x) >> 5;
  int lane = threadIdx.x & 31;
  if (row >= n) return;
  const float* a = h + (size_t)row * OUT_F;
  float v[4]; float s = 0.0f;
#pragma unroll
  for (int t = 0; t < 4; ++t) { v[t] = a[lane + t * 32]; s += v[t] * v[t]; }
#pragma unroll
  for (int off = 16; off >= 1; off >>= 1) s += __shfl_xor(s, off, 32);
  float r = rsqrtf(s);
#pragma unroll
  for (int t = 0; t < 4; ++t) z[(size_t)row * OUT_F + lane + t * 32] = v[t] * r;
}

__global__ __launch_bounds__(256)
void k_pos(const float* __restrict__ z1, const float* __restrict__ z2,
           float* __restrict__ pos, int n) {
  int row  = (blockIdx.x * blockDim.x + threadIdx.x) >> 5;
  int lane = threadIdx.x & 31;
  if (row >= n) return;
  const float* a = z1 + (size_t)row * OUT_F;
  const float* b = z2 + (size_t)row * OUT_F;
  float s = 0.0f;
#pragma unroll
  for (int t = 0; t < 4; ++t) { int j = lane + t * 32; s += a[j] * b[j]; }
#pragma unroll
  for (int off = 16; off >= 1; off >>= 1) s += __shfl_xor(s, off, 32);
  if (lane == 0) pos[row] = s * TEMP_INV;
}

// Fused neg = (z1 @ z2^T)/TEMP with diagonal mask and streaming logsumexp.
// One wave handles a 16-row strip over one 1024-column chunk, emitting a
// partial (max, sum) pair per row per chunk (never materializes 256MB neg).
// Online-softmax update is BRANCH-FREE: M=max(m,v); s=s*exp(m-M)+exp(v-M).
// exp() underflow to 0 handles both the -3e38 init and the -9e15 diag mask.
__global__ __launch_bounds__(256)
void k_neg_lse(const float* __restrict__ z1, const float* __restrict__ z2,
               float* __restrict__ partMS) {
  const int wave = (blockIdx.x * blockDim.x + threadIdx.x) >> 5;
  const int lane = threadIdx.x & 31;
  const int rowTile = wave / CHUNKS;
  const int chunk   = wave % CHUNKS;
  const int row0 = rowTile << 4;
  if (row0 >= N_NODES) return;
  const int half = lane >> 4;
  const int l    = lane & 15;

  const float* Arow = z1 + (size_t)(row0 + l) * OUT_F + 2 * half;
  float m[8], s[8];
#pragma unroll
  for (int r = 0; r < 8; ++r) { m[r] = -3.0e38f; s[r] = 0.0f; }

  const int c_beg = chunk * CHUNK_COLS;
  for (int c0 = c_beg; c0 < c_beg + CHUNK_COLS; c0 += 16) {
    v8f acc = {};
    // B[k][n] = z2[c0+n][k]  (z2 row-major, K contiguous -> float2 loads)
    const float* Brow = z2 + (size_t)(c0 + l) * OUT_F + 2 * half;
#pragma unroll 4
    for (int k0 = 0; k0 < OUT_F; k0 += 4) {
      float2 av = *(const float2*)(Arow + k0);
      float2 bv = *(const float2*)(Brow + k0);
      v2f a; a.x = av.x; a.y = av.y;
      v2f b; b.x = bv.x; b.y = bv.y;
      acc = __builtin_amdgcn_wmma_f32_16x16x4_f32(false, a, false, b,
                                                  (short)0, acc, false, false);
    }
#pragma unroll
    for (int r = 0; r < 8; ++r) {
      int row = row0 + r + 8 * half;
      int col = c0 + l;
      float v = acc[r] * TEMP_INV;
      v = (row == col) ? -9.0e15f : v;    // diagonal mask (cndmask, no branch)
      float M = fmaxf(m[r], v);
      s[r] = s[r] * __expf(m[r] - M) + __expf(v - M);
      m[r] = M;
    }
  }
  // combine (m,s) across the 16 lanes of each half-wave (branch-free)
#pragma unroll
  for (int off = 8; off >= 1; off >>= 1) {
#pragma unroll
    for (int r = 0; r < 8; ++r) {
      float mo = __shfl_xor(m[r], off, 32);
      float so = __shfl_xor(s[r], off, 32);
      float M  = fmaxf(m[r], mo);
      s[r] = s[r] * __expf(m[r] - M) + so * __expf(mo - M);
      m[r] = M;
    }
  }
  if (l == 0) {
#pragma unroll
    for (int r = 0; r < 8; ++r) {
      int row = row0 + r + 8 * half;
      partMS[((size_t)row * CHUNKS + chunk) * 2 + 0] = m[r];
      partMS[((size_t)row * CHUNKS + chunk) * 2 + 1] = s[r];
    }
  }
}

__global__ void k_lse_finalize(const float* __restrict__ partMS, float* __restrict__ lse, int n) {
  int i = blockIdx.x * blockDim.x + threadIdx.x;
  if (i >= n) return;
  float M = -3.0e38f;
#pragma unroll
  for (int c = 0; c < CHUNKS; ++c) M = fmaxf(M, partMS[((size_t)i * CHUNKS + c) * 2]);
  float S = 0.0f;
#pragma unroll
  for (int c = 0; c < CHUNKS; ++c)
    S += partMS[((size_t)i * CHUNKS + c) * 2 + 1] *
         __expf(partMS[((size_t)i * CHUNKS + c) * 2] - M);
  lse[i] = M + __logf(S);
}

__global__ void k_lossc_sum(const float* __restrict__ pos, const float* __restrict__ lse,
                            float* __restrict__ accum, int n) {
  __shared__ float sh[256];
  int tid = threadIdx.x;
  long i = (long)blockIdx.x * blockDim.x + tid;
  long stride = (long)gridDim.x * blockDim.x;
  float s = 0.0f;
  for (; i < n; i += stride) s += pos[i] - lse[i];
  sh[tid] = s; __syncthreads();
  for (int o = 128; o > 0; o >>= 1) { if (tid < o) sh[tid] += sh[tid + o]; __syncthreads(); }
  if (tid == 0) atomicAdd(accum, sh[0]);
}

// =====================================================================
// Diffusion pieces
// =====================================================================
// scal layout: [0]=loss_c_sum [1]=mse_sum [2]=ac [3]=sqrt(ac) [4]=sqrt(1-ac) [5]=tv
__global__ void k_diff_scalars(const int* __restrict__ tptr, float* __restrict__ scal) {
  if (blockIdx.x == 0 && threadIdx.x == 0) {
    int t = tptr[0];
    float ac = 1.0f;
    for (int i = 0; i <= t; ++i) {
      float beta = 1e-4f + (0.02f - 1e-4f) * (float)i / (float)(T_DIFF - 1);
      ac *= (1.0f - beta);
    }
    scal[2] = ac;
    scal[3] = sqrtf(ac);
    scal[4] = sqrtf(1.0f - ac);
    scal[5] = (float)t / (float)T_DIFF;
  }
}

__global__ void k_te(const float* __restrict__ tm1w, const float* __restrict__ tm1b,
                     const float* __restrict__ tm2w, const float* __restrict__ tm2b,
                     const float* __restrict__ scal, float* __restrict__ te) {
  __shared__ float t1[OUT_F];
  int j = threadIdx.x;
  float tv = scal[5];
  t1[j] = fmaxf(tv * tm1w[j] + tm1b[j], 0.0f);
  __syncthreads();
  float s = tm2b[j];
  for (int i = 0; i < OUT_F; ++i) s += t1[i] * tm2w[i * OUT_F + j];
  te[j] = s;
}

__global__ void k_hnoisy(const float* __restrict__ hx, const float* __restrict__ noise,
                         const float* __restrict__ te, const float* __restrict__ scal,
                         float* __restrict__ h, long n) {
  long i = (long)blockIdx.x * blockDim.x + threadIdx.x;
  if (i < n) {
    int j = (int)(i & (OUT_F - 1));
    h[i] = scal[3] * hx[i] + scal[4] * noise[i] + te[j];
  }
}

__global__ void k_mse(const float* __restrict__ a, const float* __restrict__ b,
                      float* __restrict__ accum, long n) {
  __shared__ float sh[256];
  int tid = threadIdx.x;
  long i = (long)blockIdx.x * blockDim.x + tid;
  long stride = (long)gridDim.x * blockDim.x;
  float s = 0.0f;
  for (; i < n; i += stride) { float d = a[i] - b[i]; s += d * d; }
  sh[tid] = s; __syncthreads();
  for (int o = 128; o > 0; o >>= 1) { if (tid < o) sh[tid] += sh[tid + o]; __syncthreads(); }
  if (tid == 0) atomicAdd(accum, sh[0]);
}

__global__ void k_final(const float* __restrict__ scal, float* __restrict__ out) {
  if (blockIdx.x == 0 && threadIdx.x == 0) {
    float lc = -scal[0] / (float)N_NODES;
    float lg =  scal[1] / (float)((long)N_NODES * OUT_F);
    out[0] = GAMMA * lc + (1.0f - GAMMA) * lg;
  }
}

// =====================================================================
// Host-side layer helpers (all launches on `stream`)
// =====================================================================
static inline int blocksFor(long n, int bs) { return (int)((n + bs - 1) / bs); }

template<int F>
static void gcn_layer(const float* Ain, int K, const float* Wp, const float* bp,
                      const float* g, const float* be, bool ln_relu,
                      float* xw, float* agg, float* out,
                      const int* src, const int* dst, const float* coef,
                      const float* dinv, hipStream_t s) {
  int waves = (N_NODES / 16) * (F / 16);
  k_gemm_wmma<0, 0><<<waves / 8, 256, 0, s>>>(Ain, Wp, (const float*)nullptr, xw,
                                              N_NODES, K, F);
  k_zero<<<2048, 256, 0, s>>>(agg, (long)N_NODES * F);
  long th = (long)N_EDGES * (F / 4);
  k_scatter_gcn<<<blocksFor(th, 256), 256, 0, s>>>(xw, src, dst, coef, agg, N_EDGES, F);
  if (ln_relu)
    k_gcn_post<F, 1><<<N_NODES / 8, 256, 0, s>>>(agg, xw, dinv, bp, g, be, out, N_NODES);
  else
    k_gcn_post<F, 0><<<N_NODES / 8, 256, 0, s>>>(agg, xw, dinv, bp, g, be, out, N_NODES);
}

static void hgc_layer(const float* Ain, int K, const float* Wp, const float* bp,
                      const float* g, const float* be,
                      float* xw, float* he, float* agg, float* out,
                      const int* nidx, const int* eidx, const float* Binv,
                      const float* Dinv, hipStream_t s) {
  int waves = (N_NODES / 16) * (HID_F / 16);
  k_gemm_wmma<0, 0><<<waves / 8, 256, 0, s>>>(Ain, Wp, (const float*)nullptr, xw,
                                              N_NODES, K, HID_F);
  long th = (long)N_EDGES * (HID_F / 4);
  k_zero<<<2048, 256, 0, s>>>(he, (long)N_NODES * HID_F);
  k_scatter_n2e<<<blocksFor(th, 256), 256, 0, s>>>(xw, nidx, eidx, he, N_EDGES, HID_F);
  k_zero<<<2048, 256, 0, s>>>(agg, (long)N_NODES * HID_F);
  k_scatter_e2n<<<blocksFor(th, 256), 256, 0, s>>>(he, nidx, eidx, Binv, agg, N_EDGES, HID_F);
  k_hgc_post<HID_F><<<N_NODES / 8, 256, 0, s>>>(agg, Dinv, bp, g, be, out, N_NODES);
}

// =====================================================================
// Entry point
// =====================================================================
extern "C" void kernel_launch(void* const* d_in, const int* in_sizes, int n_in,
                              void* d_out, int out_size, void* d_ws, size_t ws_size,
                              hipStream_t stream) {
  (void)in_sizes; (void)n_in; (void)out_size; (void)ws_size;
  const float* x      = (const float*)d_in[0];
  const int*   ei     = (const int*)d_in[1];   // [2, E]
  const int*   hei    = (const int*)d_in[2];   // [2, E]
  const float* feat_u = (const float*)d_in[3];
  const float* edge_u = (const float*)d_in[4];
  const float* noise  = (const float*)d_in[5];
  const int*   tptr   = (const int*)d_in[6];
  const float* P[36];
  for (int i = 0; i < 36; ++i) P[i] = (const float*)d_in[7 + i];
  // P: 0 ex_gcn1_w 1 ex_gcn1_b 2 ex_gcn2_w 3 ex_gcn2_b 4 ex_ln1_g 5 ex_ln1_b
  //    6 ex_p1_w 7 ex_p1_b 8 ex_p2_w 9 ex_p2_b
  //    10 ey_gcn1_w 11 ey_gcn1_b 12 ey_gcn2_w 13 ey_gcn2_b 14 ey_ln1_g 15 ey_ln1_b
  //    16 ey_hgc1_w 17 ey_hgc1_b 18 ey_hgc2_w 19 ey_hgc2_b 20 ey_ln2_g 21 ey_ln2_b
  //    22 ey_p1_w 23 ey_p1_b 24 ey_p2_w 25 ey_p2_b
  //    26 dn_tm1_w 27 dn_tm1_b 28 dn_tm2_w 29 dn_tm2_b
  //    30 dn_c1_w 31 dn_c1_b 32 dn_c2_w 33 dn_c2_b 34 dn_ln_g 35 dn_ln_b
  const int* src  = ei;
  const int* dst  = ei + N_EDGES;
  const int* nidx = hei;
  const int* eidx = hei + N_EDGES;

  // ------- workspace layout (floats) -------
  float* w = (float*)d_ws;
  float* x_bar  = w; w += (long)N_NODES * IN_F;
  float* ew     = w; w += N_EDGES;
  float* coef   = w; w += N_EDGES;
  float* deg    = w; w += N_NODES;
  float* dinv   = w; w += N_NODES;
  float* Binv   = w; w += N_NODES;   // counts -> reciprocal in place
  float* Dinv   = w; w += N_NODES;
  float* xw     = w; w += (long)N_NODES * HID_F;
  float* agg    = w; w += (long)N_NODES * HID_F;
  float* h1     = w; w += (long)N_NODES * HID_F;
  float* h2     = w; w += (long)N_NODES * HID_F;
  float* he     = w; w += (long)N_NODES * HID_F;
  float* hx     = w; w += (long)N_NODES * OUT_F;
  float* hy     = w; w += (long)N_NODES * OUT_F;
  float* z1     = w; w += (long)N_NODES * OUT_F;
  float* z2     = w; w += (long)N_NODES * OUT_F;
  float* tmpO   = w; w += (long)N_NODES * OUT_F;
  float* hb     = w; w += (long)N_NODES * OUT_F;
  float* hh     = w; w += (long)N_NODES * OUT_F;
  float* partMS = w; w += (long)N_NODES * CHUNKS * 2;
  float* pos    = w; w += N_NODES;
  float* lse    = w; w += N_NODES;
  float* te     = w; w += OUT_F;
  float* scal   = w; w += 16;

  // ------- setup / augmentation -------
  k_zero<<<1, 32, 0, stream>>>(scal, 16);
  k_zero<<<blocksFor(N_NODES, 256), 256, 0, stream>>>(deg, N_NODES);
  k_zero<<<blocksFor(N_NODES, 256), 256, 0, stream>>>(Binv, N_NODES);
  k_zero<<<blocksFor(N_NODES, 256), 256, 0, stream>>>(Dinv, N_NODES);
  k_mask_x<<<blocksFor((long)N_NODES * IN_F, 256), 256, 0, stream>>>(x, feat_u, x_bar,
                                                                     (long)N_NODES * IN_F);
  k_edge_w<<<blocksFor(N_EDGES, 256), 256, 0, stream>>>(edge_u, ew, N_EDGES);
  k_deg_gcn<<<blocksFor(N_EDGES, 256), 256, 0, stream>>>(dst, ew, deg, N_EDGES);
  k_dinv<<<blocksFor(N_NODES, 256), 256, 0, stream>>>(deg, dinv, N_NODES);
  k_coef<<<blocksFor(N_EDGES, 256), 256, 0, stream>>>(src, dst, ew, dinv, coef, N_EDGES);
  k_deg_hyper<<<blocksFor(N_EDGES, 256), 256, 0, stream>>>(nidx, eidx, Dinv, Binv, N_EDGES);
  k_recip<<<blocksFor(N_NODES, 256), 256, 0, stream>>>(Binv, N_NODES);
  k_recip<<<blocksFor(N_NODES, 256), 256, 0, stream>>>(Dinv, N_NODES);

  // ------- encoder_x -------
  gcn_layer<HID_F>(x_bar, IN_F,  P[0], P[1], P[4], P[5], true, xw, agg, h1,
                   src, dst, coef, dinv, stream);
  gcn_layer<HID_F>(h1,    HID_F, P[2], P[3], P[4], P[5], true, xw, agg, h1,
                   src, dst, coef, dinv, stream);
  k_gemm_wmma<1, 1><<<(N_NODES / 16) * (OUT_F / 16) / 8, 256, 0, stream>>>(
      h1, P[6], P[7], tmpO, N_NODES, HID_F, OUT_F);
  k_gemm_wmma<1, 0><<<(N_NODES / 16) * (OUT_F / 16) / 8, 256, 0, stream>>>(
      tmpO, P[8], P[9], hx, N_NODES, OUT_F, OUT_F);

  // ------- encoder_y (GCN branch) -------
  gcn_layer<HID_F>(x_bar, IN_F,  P[10], P[11], P[14], P[15], true, xw, agg, h1,
                   src, dst, coef, dinv, stream);
  gcn_layer<HID_F>(h1,    HID_F, P[12], P[13], P[14], P[15], true, xw, agg, h1,
                   src, dst, coef, dinv, stream);
  // ------- encoder_y (hypergraph branch) -------
  hgc_layer(x_bar, IN_F,  P[16], P[17], P[20], P[21], xw, he, agg, h2,
            nidx, eidx, Binv, Dinv, stream);
  hgc_layer(h2,    HID_F, P[18], P[19], P[20], P[21], xw, he, agg, h2,
            nidx, eidx, Binv, Dinv, stream);
  k_avg<<<blocksFor((long)N_NODES * HID_F, 256), 256, 0, stream>>>(h1, h2, h1,
                                                                   (long)N_NODES * HID_F);
  k_gemm_wmma<1, 1><<<(N_NODES / 16) * (OUT_F / 16) / 8, 256, 0, stream>>>(
      h1, P[22], P[23], tmpO, N_NODES, HID_F, OUT_F);
  k_gemm_wmma<1, 0><<<(N_NODES / 16) * (OUT_F / 16) / 8, 256, 0, stream>>>(
      tmpO, P[24], P[25], hy, N_NODES, OUT_F, OUT_F);

  // ------- contrastive loss -------
  k_rownorm<<<N_NODES / 8, 256, 0, stream>>>(hx, z1, N_NODES);
  k_rownorm<<<N_NODES / 8, 256, 0, stream>>>(hy, z2, N_NODES);
  k_pos<<<N_NODES / 8, 256, 0, stream>>>(z1, z2, pos, N_NODES);
  k_neg_lse<<<(N_NODES / 16) * CHUNKS / 8, 256, 0, stream>>>(z1, z2, partMS);
  k_lse_finalize<<<blocksFor(N_NODES, 256), 256, 0, stream>>>(partMS, lse, N_NODES);
  k_lossc_sum<<<32, 256, 0, stream>>>(pos, lse, scal + 0, N_NODES);

  // ------- diffusion denoiser -------
  k_diff_scalars<<<1, 32, 0, stream>>>(tptr, scal);
  k_te<<<1, OUT_F, 0, stream>>>(P[26], P[27], P[28], P[29], scal, te);
  k_hnoisy<<<blocksFor((long)N_NODES * OUT_F, 256), 256, 0, stream>>>(
      hx, noise, te, scal, hb, (long)N_NODES * OUT_F);
  gcn_layer<OUT_F>(hb, OUT_F, P[30], P[31], P[34], P[35], true,  xw, agg, hb,
                   src, dst, coef, dinv, stream);
  gcn_layer<OUT_F>(hb, OUT_F, P[32], P[33], P[34], P[35], false, xw, agg, hh,
                   src, dst, coef, dinv, stream);
  k_mse<<<256, 256, 0, stream>>>(hh, hx, scal + 1, (long)N_NODES * OUT_F);

  // ------- outputs: [loss, h_x] -------
  float* out = (float*)d_out;
  k_final<<<1, 32, 0, stream>>>(scal, out);
  k_copy<<<blocksFor((long)N_NODES * OUT_F, 256), 256, 0, stream>>>(
      hx, out + 1, (long)N_NODES * OUT_F);
}